// GeoKNO_67705864454663
// MI455X (gfx1250) — compile-verified
//
#include <hip/hip_runtime.h>
#include <math.h>

// ---------------------------------------------------------------------------
// GeoKNO forward for MI455X (gfx1250), fp32 WMMA path, fully deterministic.
// NOTE: x3 (edge gradient branch) is dead in the reference (h = x1 + x2), so
// directed_edges / edge_gradient_weights / gw_* are intentionally unused.
// ---------------------------------------------------------------------------

typedef float v2f __attribute__((ext_vector_type(2)));
typedef float v8f __attribute__((ext_vector_type(8)));

#define B_   2
#define N_   40000
#define C_   128
#define K_   64
#define L_   4
#define FC_  128
#define IN_  3

#define FW_NSLICE   125
#define FW_SLICELEN (N_ / FW_NSLICE)   // 320

__device__ __forceinline__ float gelu_exact(float x) {
    return 0.5f * x * (1.0f + erff(x * 0.70710678118654752f));
}

__device__ __forceinline__ v8f wmma_f32(v2f a, v2f b, v8f c) {
    // D = A(16x4 f32) * B(4x16 f32) + C(16x16 f32)
    return __builtin_amdgcn_wmma_f32_16x16x4_f32(false, a, false, b, (short)0, c,
                                                 false, false);
}

// ---------------------------------------------------------------------------
// Basis: temp = nodes . modes^T ; bc/bs ([B,K,N] for the inverse side),
// weighted wbc / -wbs ([B,N,K] for the forward side), wb0 = mask*nw.
// ---------------------------------------------------------------------------
__global__ void geokno_basis_kernel(const float* __restrict__ nodes,   // [B,N,2]
                                    const float* __restrict__ modes,   // [K,2]
                                    const float* __restrict__ mask,    // [B,N]
                                    const float* __restrict__ nw,      // [B,N]
                                    float* __restrict__ bc,            // [B,K,N]
                                    float* __restrict__ bs,            // [B,K,N]
                                    float* __restrict__ wbc,           // [B,N,K]
                                    float* __restrict__ wbsm,          // [B,N,K] (= -nw*bs)
                                    float* __restrict__ wb0)           // [B,N]
{
    int idx = blockIdx.x * blockDim.x + threadIdx.x;
    if (idx >= B_ * N_ * K_) return;
    int k = idx % K_;
    int n = (idx / K_) % N_;
    int b = idx / (K_ * N_);
    float t = nodes[(b * N_ + n) * 2 + 0] * modes[k * 2 + 0]
            + nodes[(b * N_ + n) * 2 + 1] * modes[k * 2 + 1];
    float m = mask[b * N_ + n];
    float w = nw[b * N_ + n];
    float c = cosf(t) * m;
    float s = sinf(t) * m;
    bc[((size_t)b * K_ + k) * N_ + n]   = c;
    bs[((size_t)b * K_ + k) * N_ + n]   = s;
    wbc[((size_t)b * N_ + n) * K_ + k]  = c * w;
    wbsm[((size_t)b * N_ + n) * K_ + k] = -s * w;
    if (k == 0) wb0[b * N_ + n] = m * w;
}

// ---------------------------------------------------------------------------
// Lift: h[b,c,n] = sum_d x[b,n,d]*fc0_w[d,c] + fc0_b[c]
// ---------------------------------------------------------------------------
__global__ void geokno_lift_kernel(const float* __restrict__ x,     // [B,N,3]
                                   const float* __restrict__ w,     // [3,C]
                                   const float* __restrict__ bias,  // [C]
                                   float* __restrict__ h)           // [B,C,N]
{
    int idx = blockIdx.x * blockDim.x + threadIdx.x;
    if (idx >= B_ * C_ * N_) return;
    int n = idx % N_;
    int c = (idx / N_) % C_;
    int b = idx / (N_ * C_);
    const float* xp = x + (size_t)(b * N_ + n) * IN_;
    float v = bias[c] + xp[0] * w[0 * C_ + c] + xp[1] * w[1 * C_ + c]
            + xp[2] * w[2 * C_ + c];
    h[((size_t)b * C_ + c) * N_ + n] = v;
}

// ---------------------------------------------------------------------------
// Forward transform, stage 1: one wave covers 4 c-tiles (one c-group) for one
// (b, cgroup, ktile, slice) -> B fragments (wbc / wbsm rows) feed 8 WMMAs.
// 8 v8f accumulators = 64 VGPRs: no spills, and all 4 A-row offsets
// (ct*16*N*4 <= 7.68MB) fold into 24-bit instruction offsets.
// Partials written non-atomically (deterministic two-stage sum).
// part layout: [slice][B][C][K]
// ---------------------------------------------------------------------------
__global__ void geokno_fwd_transform_kernel(const float* __restrict__ hin,   // [B,C,N]
                                            const float* __restrict__ wbc,   // [B,N,K]
                                            const float* __restrict__ wbsm,  // [B,N,K]
                                            float* __restrict__ partC,
                                            float* __restrict__ partS)
{
    int wave = (blockIdx.x * blockDim.x + threadIdx.x) >> 5;   // 0..1999
    int lane = threadIdx.x & 31;
    int sub  = lane >> 4, l16 = lane & 15;

    int sl = wave % FW_NSLICE;
    int kt = (wave / FW_NSLICE) & 3;
    int cg = (wave / (FW_NSLICE * 4)) & 1;    // c-group: c in [cg*64, cg*64+64)
    int b  = wave / (FW_NSLICE * 4 * 2);
    int k0 = kt * 16;
    int cbase = cg * 64;
    int nbeg = sl * FW_SLICELEN;

    v8f accC[4], accS[4];
#pragma unroll
    for (int ct = 0; ct < 4; ++ct) { accC[ct] = (v8f)0.0f; accS[ct] = (v8f)0.0f; }

    for (int n0 = nbeg; n0 < nbeg + FW_SLICELEN; n0 += 4) {
        int na = n0 + 2 * sub;                       // K-pair handled by this half
        const float* cp = wbc  + ((size_t)b * N_ + na) * K_ + k0 + l16;
        const float* sp = wbsm + ((size_t)b * N_ + na) * K_ + k0 + l16;
        v2f bcv; bcv.x = cp[0]; bcv.y = cp[K_];      // B[k=2*sub+{0,1}][j=l16]
        v2f bsv; bsv.x = sp[0]; bsv.y = sp[K_];
        const float* hbase = hin + ((size_t)b * C_ + cbase + l16) * N_ + na;
#pragma unroll
        for (int ct = 0; ct < 4; ++ct) {
            v2f a;                                   // A[m=l16][k=2*sub+{0,1}]
            a.x = hbase[(size_t)ct * 16 * N_];
            a.y = hbase[(size_t)ct * 16 * N_ + 1];
            accC[ct] = wmma_f32(a, bcv, accC[ct]);
            accS[ct] = wmma_f32(a, bsv, accS[ct]);
        }
    }

#pragma unroll
    for (int ct = 0; ct < 4; ++ct) {
#pragma unroll
        for (int r = 0; r < 8; ++r) {
            int c = cbase + ct * 16 + r + 8 * sub;  // C/D layout: vgpr r -> rows r / r+8
            int k = k0 + l16;
            size_t p = (((size_t)sl * B_ + b) * C_ + c) * K_ + k;
            partC[p] = accC[ct][r];
            partS[p] = accS[ct][r];
        }
    }
}

// ---------------------------------------------------------------------------
// Forward transform, stage 2: fixed-order slice reduction (deterministic).
// ---------------------------------------------------------------------------
__global__ void geokno_fwd_reduce_kernel(const float* __restrict__ partC,
                                         const float* __restrict__ partS,
                                         float* __restrict__ xc,   // [B,C,K]
                                         float* __restrict__ xs)   // [B,C,K]
{
    int idx = blockIdx.x * blockDim.x + threadIdx.x;
    if (idx >= B_ * C_ * K_) return;
    float sc = 0.0f, ss = 0.0f;
    for (int s = 0; s < FW_NSLICE; ++s) {
        size_t p = (size_t)s * (B_ * C_ * K_) + idx;
        sc += partC[p];
        ss += partS[p];
    }
    xc[idx] = sc;
    xs[idx] = ss;
}

// ---------------------------------------------------------------------------
// x0[b,c] = sum_n h[b,c,n]*wb0[b,n] : one block per (b,c), fixed reduction tree.
// ---------------------------------------------------------------------------
__global__ void geokno_x0_kernel(const float* __restrict__ hin,  // [B,C,N]
                                 const float* __restrict__ wb0,  // [B,N]
                                 float* __restrict__ x0)         // [B,C]
{
    int bcid = blockIdx.x;           // = b*C + c
    int c = bcid % C_;
    int b = bcid / C_;
    const float* hrow = hin + ((size_t)b * C_ + c) * N_;
    const float* wp   = wb0 + (size_t)b * N_;
    float acc = 0.0f;
    for (int n = threadIdx.x; n < N_; n += 256) acc += hrow[n] * wp[n];
    __shared__ float red[8];
    acc += __shfl_xor(acc, 1);
    acc += __shfl_xor(acc, 2);
    acc += __shfl_xor(acc, 4);
    acc += __shfl_xor(acc, 8);
    acc += __shfl_xor(acc, 16);
    if ((threadIdx.x & 31) == 0) red[threadIdx.x >> 5] = acc;
    __syncthreads();
    if (threadIdx.x == 0) {
        float t = 0.0f;
#pragma unroll
        for (int w = 0; w < 8; ++w) t += red[w];
        x0[bcid] = t;
    }
}

// ---------------------------------------------------------------------------
// Per-k CxC spectral mixes (tiny): fcm = 2*(xc*wc - xs*ws), fsm = -2*(xs*wc + xc*ws)
// f0m = x0 * w0.
// ---------------------------------------------------------------------------
__global__ void geokno_spec_mult_kernel(const float* __restrict__ xc,
                                        const float* __restrict__ xs,
                                        const float* __restrict__ x0,
                                        const float* __restrict__ wc,  // [C,C,K] slice
                                        const float* __restrict__ ws,  // [C,C,K]
                                        const float* __restrict__ w0,  // [C,C]
                                        float* __restrict__ fcm,       // [B,C,K]
                                        float* __restrict__ fsm,       // [B,C,K]
                                        float* __restrict__ f0m)       // [B,C]
{
    int idx = blockIdx.x * blockDim.x + threadIdx.x;
    const int total_k = B_ * C_ * K_;
    if (idx < total_k) {
        int k = idx % K_;
        int o = (idx / K_) % C_;
        int b = idx / (K_ * C_);
        float ac = 0.0f, as = 0.0f;
        for (int i = 0; i < C_; ++i) {
            float xci = xc[((size_t)b * C_ + i) * K_ + k];
            float xsi = xs[((size_t)b * C_ + i) * K_ + k];
            float wci = wc[((size_t)i * C_ + o) * K_ + k];
            float wsi = ws[((size_t)i * C_ + o) * K_ + k];
            ac += xci * wci - xsi * wsi;
            as += xsi * wci + xci * wsi;
        }
        fcm[idx] = 2.0f * ac;
        fsm[idx] = -2.0f * as;
    } else if (idx < total_k + B_ * C_) {
        int j = idx - total_k;
        int o = j % C_;
        int b = j / C_;
        float a0 = 0.0f;
        for (int i = 0; i < C_; ++i)
            a0 += x0[b * C_ + i] * w0[(size_t)i * C_ + o];
        f0m[j] = a0;
    }
}

// ---------------------------------------------------------------------------
// Layer GEMM: h_out[c,n] = gelu?( fcm.bc + fsm.bs + f0m*mask + W.h_in + w_b )
// One wave covers ALL 8 c-tiles of one n-tile: every B fragment (bc/bs/h
// columns) is loaded once and feeds 8 WMMAs -> 8x less L2 read traffic.
// 8 v8f accumulators = 64 VGPRs (no spills).
// ---------------------------------------------------------------------------
__global__ void geokno_layer_kernel(const float* __restrict__ hin,  // [B,C,N]
                                    const float* __restrict__ bc,   // [B,K,N]
                                    const float* __restrict__ bs,   // [B,K,N]
                                    const float* __restrict__ fcm,  // [B,C,K]
                                    const float* __restrict__ fsm,  // [B,C,K]
                                    const float* __restrict__ f0m,  // [B,C]
                                    const float* __restrict__ mask, // [B,N]
                                    const float* __restrict__ ww,   // [C,C] slice
                                    const float* __restrict__ wb,   // [C]
                                    float* __restrict__ hout,       // [B,C,N]
                                    int apply_gelu)
{
    int gwave = (blockIdx.x * blockDim.x + threadIdx.x) >> 5;  // 0..4999
    int lane  = threadIdx.x & 31;
    int sub   = lane >> 4, l16 = lane & 15;
    int nt = gwave % (N_ / 16);
    int b  = gwave / (N_ / 16);
    int n0 = nt * 16;

    v8f acc[8];
#pragma unroll
    for (int ct = 0; ct < 8; ++ct) acc[ct] = (v8f)0.0f;

    // Phase 1: spectral inverse, reduction over K=64
    for (int k0 = 0; k0 < K_; k0 += 4) {
        int ka = k0 + 2 * sub;
        const float* bp  = bc + ((size_t)b * K_ + ka) * N_ + n0 + l16;
        const float* bp2 = bs + ((size_t)b * K_ + ka) * N_ + n0 + l16;
        v2f bbc; bbc.x = bp[0];  bbc.y = bp[N_];
        v2f bbs; bbs.x = bp2[0]; bbs.y = bp2[N_];
        const float* apc = fcm + ((size_t)b * C_ + l16) * K_ + ka;
        const float* aps = fsm + ((size_t)b * C_ + l16) * K_ + ka;
#pragma unroll
        for (int ct = 0; ct < 8; ++ct) {
            v2f a;
            a.x = apc[(size_t)ct * 16 * K_];
            a.y = apc[(size_t)ct * 16 * K_ + 1];
            acc[ct] = wmma_f32(a, bbc, acc[ct]);
            v2f a2;
            a2.x = aps[(size_t)ct * 16 * K_];
            a2.y = aps[(size_t)ct * 16 * K_ + 1];
            acc[ct] = wmma_f32(a2, bbs, acc[ct]);
        }
    }
    // Phase 2: 1x1 conv, reduction over C=128
    for (int c2 = 0; c2 < C_; c2 += 4) {
        int ca = c2 + 2 * sub;
        const float* bp = hin + ((size_t)b * C_ + ca) * N_ + n0 + l16;
        v2f bb; bb.x = bp[0]; bb.y = bp[N_];
        const float* ap0 = ww + (size_t)l16 * C_ + ca;
#pragma unroll
        for (int ct = 0; ct < 8; ++ct) {
            v2f a;
            a.x = ap0[(size_t)ct * 16 * C_];
            a.y = ap0[(size_t)ct * 16 * C_ + 1];
            acc[ct] = wmma_f32(a, bb, acc[ct]);
        }
    }
    // Epilogue: rank-1 f0*b0 term + bias + activation
    int n = n0 + l16;
    float mk = mask[b * N_ + n];
#pragma unroll
    for (int ct = 0; ct < 8; ++ct) {
#pragma unroll
        for (int r = 0; r < 8; ++r) {
            int c = ct * 16 + r + 8 * sub;
            float v = acc[ct][r] + wb[c] + f0m[b * C_ + c] * mk;
            if (apply_gelu) v = gelu_exact(v);
            hout[((size_t)b * C_ + c) * N_ + n] = v;
        }
    }
}

// ---------------------------------------------------------------------------
// Fused MLP head: out[b,n] = gelu(h^T . fc1 + b1) . fc2 + b2
// One wave per 16 n's; 8 F-tiles in registers, GELU on accumulators,
// deterministic cross-lane reduction to the scalar output.
// ---------------------------------------------------------------------------
__global__ void geokno_mlp_kernel(const float* __restrict__ hin,  // [B,C,N]
                                  const float* __restrict__ w1,   // [C,FC]
                                  const float* __restrict__ b1,   // [FC]
                                  const float* __restrict__ w2,   // [FC]
                                  const float* __restrict__ b2,   // [1]
                                  float* __restrict__ out)        // [B,N]
{
    int gwave = (blockIdx.x * blockDim.x + threadIdx.x) >> 5;
    int lane  = threadIdx.x & 31;
    int sub   = lane >> 4, l16 = lane & 15;
    int nt = gwave % (N_ / 16);
    int b  = gwave / (N_ / 16);
    int n0 = nt * 16;

    v8f acc[FC_ / 16];
#pragma unroll
    for (int j = 0; j < FC_ / 16; ++j) acc[j] = (v8f)0.0f;

    for (int c2 = 0; c2 < C_; c2 += 4) {
        int ca = c2 + 2 * sub;
        v2f a;
        const float* ap = hin + ((size_t)b * C_ + ca) * N_ + n0 + l16;
        a.x = ap[0]; a.y = ap[N_];  // A[m=l16][k=2*sub+{0,1}] = h[ca+{0,1}][n0+l16]
#pragma unroll
        for (int j = 0; j < FC_ / 16; ++j) {
            v2f bb;
            const float* bp = w1 + (size_t)ca * FC_ + j * 16 + l16;
            bb.x = bp[0]; bb.y = bp[FC_];
            acc[j] = wmma_f32(a, bb, acc[j]);
        }
    }

    float part[8];
#pragma unroll
    for (int r = 0; r < 8; ++r) part[r] = 0.0f;
#pragma unroll
    for (int j = 0; j < FC_ / 16; ++j) {
        int f = j * 16 + l16;
        float bb1 = b1[f], ww2 = w2[f];
#pragma unroll
        for (int r = 0; r < 8; ++r)
            part[r] += gelu_exact(acc[j][r] + bb1) * ww2;
    }
#pragma unroll
    for (int r = 0; r < 8; ++r) {
        part[r] += __shfl_xor(part[r], 1);
        part[r] += __shfl_xor(part[r], 2);
        part[r] += __shfl_xor(part[r], 4);
        part[r] += __shfl_xor(part[r], 8);
    }
    if (l16 == 0) {
        float bias2 = b2[0];
#pragma unroll
        for (int r = 0; r < 8; ++r) {
            int n = n0 + r + 8 * sub;   // lanes 0 / 16 cover rows r / r+8
            out[(size_t)b * N_ + n] = part[r] + bias2;
        }
    }
}

// ---------------------------------------------------------------------------
// Host launch
// ---------------------------------------------------------------------------
extern "C" void kernel_launch(void* const* d_in, const int* in_sizes, int n_in,
                              void* d_out, int out_size, void* d_ws, size_t ws_size,
                              hipStream_t stream)
{
    const float* x     = (const float*)d_in[0];
    const float* mask  = (const float*)d_in[1];
    const float* nodes = (const float*)d_in[2];
    const float* nw    = (const float*)d_in[3];
    // d_in[4] directed_edges / d_in[5] edge_gradient_weights: dead (x3 never added)
    const float* modes = (const float*)d_in[6];
    const float* fc0_w = (const float*)d_in[7];
    const float* fc0_b = (const float*)d_in[8];
    const float* sp_wc = (const float*)d_in[9];
    const float* sp_ws = (const float*)d_in[10];
    const float* sp_w0 = (const float*)d_in[11];
    const float* w_w   = (const float*)d_in[12];
    const float* w_b   = (const float*)d_in[13];
    // d_in[14] gw_w / d_in[15] gw_b: dead
    const float* fc1_w = (const float*)d_in[16];
    const float* fc1_b = (const float*)d_in[17];
    const float* fc2_w = (const float*)d_in[18];
    const float* fc2_b = (const float*)d_in[19];
    float* out = (float*)d_out;

    // Workspace carve-up (~165 MB fp32)
    float* p    = (float*)d_ws;
    float* bcb  = p; p += (size_t)B_ * K_ * N_;
    float* bsb  = p; p += (size_t)B_ * K_ * N_;
    float* wbc  = p; p += (size_t)B_ * N_ * K_;
    float* wbsm = p; p += (size_t)B_ * N_ * K_;
    float* wb0  = p; p += (size_t)B_ * N_;
    float* hA   = p; p += (size_t)B_ * C_ * N_;
    float* hB   = p; p += (size_t)B_ * C_ * N_;
    float* xc   = p; p += (size_t)B_ * C_ * K_;
    float* xs   = p; p += (size_t)B_ * C_ * K_;
    float* x0   = p; p += (size_t)B_ * C_;
    float* fcm  = p; p += (size_t)B_ * C_ * K_;
    float* fsm  = p; p += (size_t)B_ * C_ * K_;
    float* f0m  = p; p += (size_t)B_ * C_;

    {
        int total = B_ * N_ * K_;
        geokno_basis_kernel<<<(total + 255) / 256, 256, 0, stream>>>(
            nodes, modes, mask, nw, bcb, bsb, wbc, wbsm, wb0);
    }
    {
        int total = B_ * C_ * N_;
        geokno_lift_kernel<<<(total + 255) / 256, 256, 0, stream>>>(
            x, fc0_w, fc0_b, hA);
    }

    float* hin  = hA;
    float* hout = hB;
    for (int l = 0; l < L_; ++l) {
        // Stage forward-transform partials in the (currently dead) hout buffer:
        // 2 x 125*B*C*K floats = 16.4 MB <= 41 MB.
        float* partC = hout;
        float* partS = hout + (size_t)FW_NSLICE * B_ * C_ * K_;
        {
            int waves = B_ * 2 * 4 * FW_NSLICE;          // 2000 waves
            geokno_fwd_transform_kernel<<<waves * 32 / 256, 256, 0, stream>>>(
                hin, wbc, wbsm, partC, partS);
        }
        {
            int total = B_ * C_ * K_;
            geokno_fwd_reduce_kernel<<<(total + 255) / 256, 256, 0, stream>>>(
                partC, partS, xc, xs);
        }
        geokno_x0_kernel<<<B_ * C_, 256, 0, stream>>>(hin, wb0, x0);
        {
            int total = B_ * C_ * K_ + B_ * C_;
            geokno_spec_mult_kernel<<<(total + 255) / 256, 256, 0, stream>>>(
                xc, xs, x0,
                sp_wc + (size_t)l * C_ * C_ * K_,
                sp_ws + (size_t)l * C_ * C_ * K_,
                sp_w0 + (size_t)l * C_ * C_,
                fcm, fsm, f0m);
        }
        {
            int waves = B_ * (N_ / 16);                  // 5000 waves
            geokno_layer_kernel<<<waves * 32 / 256, 256, 0, stream>>>(
                hin, bcb, bsb, fcm, fsm, f0m, mask,
                w_w + (size_t)l * C_ * C_, w_b + (size_t)l * C_,
                hout, (l != L_ - 1) ? 1 : 0);
        }
        float* t = hin; hin = hout; hout = t;
    }
    {
        int waves = B_ * (N_ / 16);                      // 5000 waves
        geokno_mlp_kernel<<<waves * 32 / 256, 256, 0, stream>>>(
            hin, fc1_w, fc1_b, fc2_w, fc2_b, out);
    }
    (void)in_sizes; (void)n_in; (void)out_size; (void)ws_size;
}